// QuickFPSFunction_38001870635079
// MI455X (gfx1250) — compile-verified
//
#include <hip/hip_runtime.h>
#include <stdint.h>

// ---------------- problem constants (match setup_inputs) ----------------
#define B_N 32
#define P_N 131072
#define S_N 512
#define WGB 8                      // workgroups (chunks) per batch
#define CHUNK (P_N / WGB)          // 16384 points per WG
#define NT 512                     // threads per WG (16 wave32s)
#define PT (CHUNK / NT)            // 32 points per thread, kept in VGPRs
#define TILE_PTS 4096
#define TILE_BYTES (TILE_PTS * 12) // 48 KB staging tile
#define NTILES (CHUNK / TILE_PTS)  // 4

typedef unsigned long long u64;
typedef unsigned int u32;
typedef int vi4 __attribute__((vector_size(16)));   // <4 x i32>, as the builtin wants

#ifndef __has_builtin
#define __has_builtin(x) 0
#endif

#define AS1 __attribute__((address_space(1)))
#define AS3 __attribute__((address_space(3)))

// ---- CDNA5 async global->LDS copy (ASYNCcnt path) ----
__device__ __forceinline__ void async_copy_b128(const void* gp, void* lp) {
#if __has_builtin(__builtin_amdgcn_global_load_async_to_lds_b128)
  __builtin_amdgcn_global_load_async_to_lds_b128((AS1 vi4*)gp, (AS3 vi4*)lp, 0, 0);
#else
  u32 loff = (u32)(uint64_t)(AS3 char*)lp;
  asm volatile("global_load_async_to_lds_b128 %0, %1, off"
               :: "v"(loff), "v"((uint64_t)gp) : "memory");
#endif
}

__device__ __forceinline__ void wait_asynccnt0() {
#if __has_builtin(__builtin_amdgcn_s_wait_asynccnt)
  __builtin_amdgcn_s_wait_asynccnt(0);
#else
  asm volatile("s_wait_asynccnt 0" ::: "memory");
#endif
}

// 64-bit xor-shuffle built from two 32-bit shuffles (wave32)
__device__ __forceinline__ u64 shfl_xor_u64(u64 v, int m) {
  u32 lo = __shfl_xor((u32)v, m, 32);
  u32 hi = __shfl_xor((u32)(v >> 32), m, 32);
  return ((u64)hi << 32) | lo;
}

__global__ __launch_bounds__(NT)
void fps_kernel(const float* __restrict__ points,
                int* __restrict__ outIdx, float* __restrict__ outPts,
                u64* __restrict__ slots, u32* __restrict__ arrivals)
{
  const int tid   = threadIdx.x;
  const int b     = blockIdx.x >> 3;         // batch
  const int chunk = blockIdx.x & (WGB - 1);  // chunk within batch

  const float* bpts = points + (size_t)b * P_N * 3;       // this batch, AoS xyz
  const float* cpts = bpts + (size_t)chunk * CHUNK * 3;   // this chunk

  __shared__ __align__(16) char stage[TILE_BYTES];
  __shared__ u64 waveBest[NT / 32];
  __shared__ float cbc[3];

  float px[PT], py[PT], pz[PT], dmin[PT];

  // ---------- prologue: DMA points chunk -> LDS tiles -> VGPRs ----------
  const int myTile = tid >> 7;   // 128 threads (4 waves) consume each tile
  const int lt     = tid & 127;
  for (int t = 0; t < NTILES; ++t) {
    const char* gsrc = (const char*)cpts + (size_t)t * TILE_BYTES;
#pragma unroll
    for (int j = 0; j < TILE_BYTES / 16 / NT; ++j) {      // 6 x b128 per thread
      int off = (tid + j * NT) * 16;
      async_copy_b128(gsrc + off, stage + off);
    }
    wait_asynccnt0();
    __syncthreads();
    if (myTile == t) {
      const char* base = stage + (size_t)lt * PT * 12;
#pragma unroll
      for (int k = 0; k < PT; ++k) {
        px[k] = *(const float*)(base + k * 12 + 0);
        py[k] = *(const float*)(base + k * 12 + 4);
        pz[k] = *(const float*)(base + k * 12 + 8);
      }
    }
    __syncthreads();  // tile fully consumed before next DMA overwrites it
  }

  // ---------- round 0: first sample is always index 0 ----------
  if (tid == 0) {
    float x0 = bpts[0], y0 = bpts[1], z0 = bpts[2];
    cbc[0] = x0; cbc[1] = y0; cbc[2] = z0;
    if (chunk == 0) {
      outIdx[b * S_N] = 0;
      float* o = outPts + (size_t)b * S_N * 3;
      o[0] = x0; o[1] = y0; o[2] = z0;
    }
  }
  __syncthreads();
  float cx = cbc[0], cy = cbc[1], cz = cbc[2];

  const float INF = __int_as_float(0x7f800000);
#pragma unroll
  for (int k = 0; k < PT; ++k) dmin[k] = INF;

  const u32 gbase = (u32)(chunk * CHUNK + tid * PT);  // global point index base
  u64* slot = slots + (size_t)b * S_N;
  u32* arr  = arrivals + b * 16;                      // 64B-strided counters

  // ---------- 511 sequential FPS rounds, all data in VGPRs ----------
  for (int s = 1; s < S_N; ++s) {
    u64 best = 0ull;
#pragma unroll
    for (int k = 0; k < PT; ++k) {
      float dx = px[k] - cx, dy = py[k] - cy, dz = pz[k] - cz;
      float dd = dx * dx + dy * dy + dz * dz;
      float dk = fminf(dmin[k], dd);
      dmin[k] = dk;
      // pack: high 32 = dist bits (monotonic for non-negative floats),
      // low 32 = ~idx so equal dists resolve to the LOWEST index (jnp.argmax tie rule)
      u64 pk = ((u64)__float_as_uint(dk) << 32) | (u32)(~(gbase + (u32)k));
      best = best > pk ? best : pk;
    }
    // wave32 reduction
#pragma unroll
    for (int off = 16; off >= 1; off >>= 1) {
      u64 o = shfl_xor_u64(best, off);
      best = best > o ? best : o;
    }
    if ((tid & 31) == 0) waveBest[tid >> 5] = best;
    __syncthreads();

    if (tid == 0) {
      u64 m = waveBest[0];
#pragma unroll
      for (int w = 1; w < NT / 32; ++w) { u64 v = waveBest[w]; m = m > v ? m : v; }

      atomicMax(slot + s, m);   // order-independent combine across 8 chunk-WGs
      __hip_atomic_fetch_add(arr, 1u, __ATOMIC_RELEASE, __HIP_MEMORY_SCOPE_AGENT);
      const u32 target = (u32)(WGB * s);  // monotonic arrival count
      while (__hip_atomic_load(arr, __ATOMIC_ACQUIRE, __HIP_MEMORY_SCOPE_AGENT) < target)
        __builtin_amdgcn_s_sleep(2);

      u64 win = __hip_atomic_load(slot + s, __ATOMIC_RELAXED, __HIP_MEMORY_SCOPE_AGENT);
      int widx = (int)(~(u32)win);
      const float* wp = bpts + (size_t)widx * 3;  // one L2-resident read
      float wx = wp[0], wy = wp[1], wz = wp[2];
      cbc[0] = wx; cbc[1] = wy; cbc[2] = wz;
      if (chunk == 0) {
        outIdx[b * S_N + s] = widx;
        float* o = outPts + ((size_t)b * S_N + s) * 3;
        o[0] = wx; o[1] = wy; o[2] = wz;
      }
    }
    __syncthreads();
    cx = cbc[0]; cy = cbc[1]; cz = cbc[2];
  }
}

extern "C" void kernel_launch(void* const* d_in, const int* in_sizes, int n_in,
                              void* d_out, int out_size, void* d_ws, size_t ws_size,
                              hipStream_t stream) {
  (void)in_sizes; (void)n_in; (void)out_size; (void)ws_size;
  const float* points = (const float*)d_in[0];

  // d_out: [B*S] int32 indices, then [B*S*3] float32 gathered points
  int*   outIdx = (int*)d_out;
  float* outPts = (float*)d_out + B_N * S_N;

  // d_ws layout: u64 slots[B][S] (128 KB) then 32 arrival counters @64B stride
  u64* slots    = (u64*)d_ws;
  u32* arrivals = (u32*)((char*)d_ws + (size_t)B_N * S_N * sizeof(u64));
  size_t zbytes = (size_t)B_N * S_N * sizeof(u64) + (size_t)B_N * 64;
  (void)hipMemsetAsync(d_ws, 0, zbytes, stream);   // graph-capture safe

  fps_kernel<<<dim3(B_N * WGB), dim3(NT), 0, stream>>>(points, outIdx, outPts,
                                                       slots, arrivals);
}